// GeoEdgeConv_38113539785409
// MI455X (gfx1250) — compile-verified
//
#include <hip/hip_runtime.h>
#include <math.h>

#define Bb 4
#define Cc 64
#define Nn 4096
#define Kk 20
#define NK (Nn*Kk)          // 81920 positions per batch
#define GN_EPS 1e-5f

typedef __attribute__((ext_vector_type(2))) float v2f;
typedef __attribute__((ext_vector_type(8))) float v8f;

__device__ __forceinline__ float mishf(float x) {
    float sp = (x > 20.f) ? x : log1pf(__expf(x));
    return x * tanhf(sp);
}
__device__ __forceinline__ float gnact(float x, float2 st, float g, float b) {
    return mishf((x - st.x) * st.y * g + b);
}

// Branchless sorted-descending insert: 1 cmp + 4 cndmask per level.
__device__ __forceinline__ void topk_insert(float (&vals)[Kk], int (&ids)[Kk],
                                            float cv, int ci) {
#pragma unroll
    for (int t = 0; t < Kk; ++t) {
        bool  sw = cv > vals[t];
        float nv = sw ? vals[t] : cv;
        int   ni = sw ? ids[t]  : ci;
        vals[t]  = sw ? cv : vals[t];
        ids[t]   = sw ? ci : ids[t];
        cv = nv; ci = ni;
    }
}

// ---------------------------------------------------------------- sq = |pos|^2
__global__ void sq_kernel(const float* __restrict__ pos, float* __restrict__ sq) {
    int t = blockIdx.x * blockDim.x + threadIdx.x;
    if (t >= Bb * Nn) return;
    int b = t / Nn, n = t % Nn;
    const float* p = pos + (size_t)b * 3 * Nn;
    float x = p[n], y = p[Nn + n], z = p[2 * Nn + n];
    sq[t] = x * x + y * y + z * z;
}

// ------------------------------------------------- KNN via WMMA f32 16x16x4
// One wave per 16 query rows. negd = 2*dot - sq_i - sq_j.
// Selection: lane r and lane r+16 each own half the columns of row r; the two
// sorted 20-lists are merged through LDS at the end.
__global__ void __launch_bounds__(32) knn_kernel(const float* __restrict__ pos,
                                                 const float* __restrict__ sq,
                                                 int* __restrict__ idx,
                                                 float* __restrict__ dist_sum) {
    int blk  = blockIdx.x;
    int b    = blk / (Nn / 16);
    int q0   = (blk % (Nn / 16)) * 16;
    int lane = threadIdx.x;
    int lo   = lane & 15;
    bool hi  = lane >= 16;
    int  h   = lane >> 4;
    const float* px = pos + (size_t)b * 3 * Nn;

    // A fragment: query points, M=lo ; K components {x,y}/{z,0} split by lane half
    v2f af;
    af.x = hi ? px[2 * Nn + q0 + lo] : px[q0 + lo];
    af.y = hi ? 0.f : px[Nn + q0 + lo];

    float sqq[8];
#pragma unroll
    for (int v = 0; v < 8; ++v) sqq[v] = sq[b * Nn + q0 + v + (hi ? 8 : 0)];

    float vals[Kk]; int ids[Kk];
#pragma unroll
    for (int t = 0; t < Kk; ++t) { vals[t] = -3.0e38f; ids[t] = 0; }

    __shared__ float tile[256];
    __shared__ float mv[16 * Kk];
    __shared__ int   mi[16 * Kk];

    for (int c0 = 0; c0 < Nn; c0 += 16) {
        v2f bf;                         // B fragment: key points, col = lo
        bf.x = hi ? px[2 * Nn + c0 + lo] : px[c0 + lo];
        bf.y = hi ? 0.f : px[Nn + c0 + lo];
        v8f acc = {};
        acc = __builtin_amdgcn_wmma_f32_16x16x4_f32(false, af, false, bf,
                                                    (short)0, acc, false, false);
        float sqc = sq[b * Nn + c0 + lo];
#pragma unroll
        for (int v = 0; v < 8; ++v) {
            int row = v + (hi ? 8 : 0);
            tile[row * 16 + lo] = 2.f * acc[v] - sqq[v] - sqc;
        }
        __syncthreads();
        // lane owns row lo, columns [h*8, h*8+8) of this tile
        for (int jj = 0; jj < 8; ++jj) {
            int j = h * 8 + jj;
            float vv = tile[lo * 16 + j];
            if (vv > vals[Kk - 1]) topk_insert(vals, ids, vv, c0 + j);
        }
        __syncthreads();
    }

    // merge the two half-lists of each row
    if (h == 1) {
#pragma unroll
        for (int t = 0; t < Kk; ++t) { mv[lo * Kk + t] = vals[t]; mi[lo * Kk + t] = ids[t]; }
    }
    __syncthreads();
    if (h == 0) {
        for (int t = 0; t < Kk; ++t) {
            float vv = mv[lo * Kk + t];
            if (vv <= vals[Kk - 1]) break;   // incoming list sorted descending
            topk_insert(vals, ids, vv, mi[lo * Kk + t]);
        }
        int n = q0 + lo;
        float ds = 0.f;
#pragma unroll
        for (int t = 0; t < Kk; ++t) {
            idx[((size_t)b * Nn + n) * Kk + t] = ids[t];
            ds += sqrtf(fmaxf(-vals[t], 0.f));
        }
        dist_sum[b * Nn + n] = ds;
    }
}

// -------------------------- edge features + grouped conv1 (128->32), conv2 (12->32)
__global__ void edge_kernel(const float* __restrict__ x, const float* __restrict__ pos,
                            const int* __restrict__ idx,
                            const float* __restrict__ w1, const float* __restrict__ b1,
                            const float* __restrict__ w2, const float* __restrict__ b2,
                            float* __restrict__ A1, float* __restrict__ P1) {
    int t = blockIdx.x * blockDim.x + threadIdx.x;
    if (t >= Bb * Nn * Kk) return;
    int k = t % Kk; int n = (t / Kk) % Nn; int b = t / (Nn * Kk);
    int j = idx[t];
    const float* xb = x + (size_t)b * Cc * Nn;

    float acc[32];
#pragma unroll
    for (int o = 0; o < 32; ++o) acc[o] = b1[o];
    for (int i = 0; i < 32; ++i) {
        float cfa = xb[(size_t)i * Nn + n];
        float nfa = xb[(size_t)i * Nn + j];
        float cfb = xb[(size_t)(32 + i) * Nn + n];
        float nfb = xb[(size_t)(32 + i) * Nn + j];
        float e0 = nfa - cfa, e1 = nfb - cfb;
#pragma unroll
        for (int o = 0; o < 8; ++o) {
            acc[o]      = fmaf(w1[o * 32 + i],        e0,  acc[o]);
            acc[8 + o]  = fmaf(w1[(8 + o) * 32 + i],  e1,  acc[8 + o]);
            acc[16 + o] = fmaf(w1[(16 + o) * 32 + i], cfa, acc[16 + o]);
            acc[24 + o] = fmaf(w1[(24 + o) * 32 + i], cfb, acc[24 + o]);
        }
    }
    size_t base = (size_t)b * 32 * NK + (size_t)n * Kk + k;
#pragma unroll
    for (int o = 0; o < 32; ++o) A1[base + (size_t)o * NK] = acc[o];

    const float* pb = pos + (size_t)b * 3 * Nn;
    float pcx = pb[n], pcy = pb[Nn + n], pcz = pb[2 * Nn + n];
    float pnx = pb[j], pny = pb[Nn + j], pnz = pb[2 * Nn + j];
    float dx = pnx - pcx, dy = pny - pcy, dz = pnz - pcz;
    float fin[12] = { pcx, pcy, pcz, pnx, pny, pnz, dx, dy, dz, dx*dx, dy*dy, dz*dz };
#pragma unroll
    for (int o = 0; o < 32; ++o) {
        int g = o >> 3;
        float a = b2[o];
        a = fmaf(w2[o * 3 + 0], fin[g * 3 + 0], a);
        a = fmaf(w2[o * 3 + 1], fin[g * 3 + 1], a);
        a = fmaf(w2[o * 3 + 2], fin[g * 3 + 2], a);
        P1[base + (size_t)o * NK] = a;
    }
}

// --------------------------------------------- per-group mean / rstd reduction
// L must be a multiple of 4; region bases are 256B aligned.
__global__ void stats_kernel(const float* __restrict__ buf, float2* __restrict__ out, int L) {
    const float4* p = (const float4*)(buf + (size_t)blockIdx.x * L);
    int L4 = L >> 2;
    float s = 0.f, s2 = 0.f;
    for (int i = threadIdx.x; i < L4; i += 256) {
        float4 v = p[i];
        s += (v.x + v.y) + (v.z + v.w);
        s2 = fmaf(v.x, v.x, fmaf(v.y, v.y, fmaf(v.z, v.z, fmaf(v.w, v.w, s2))));
    }
    __shared__ float sh[256], sh2[256];
    sh[threadIdx.x] = s; sh2[threadIdx.x] = s2;
    __syncthreads();
    for (int w = 128; w > 0; w >>= 1) {
        if (threadIdx.x < w) { sh[threadIdx.x] += sh[threadIdx.x + w]; sh2[threadIdx.x] += sh2[threadIdx.x + w]; }
        __syncthreads();
    }
    if (threadIdx.x == 0) {
        float m = sh[0] / (float)L;
        float var = sh2[0] / (float)L - m * m;
        out[blockIdx.x] = make_float2(m, rsqrtf(var + GN_EPS));
    }
}

// ----------------- fused GN+mish -> 32x32 GEMM (+bias), in place, WMMA f32
__global__ void __launch_bounds__(256) gemm32_kernel(float* __restrict__ buf,
        const float2* __restrict__ st, const float* __restrict__ gamma,
        const float* __restrict__ beta, const float* __restrict__ W,
        const float* __restrict__ bias) {
    int wave = (blockIdx.x * 256 + threadIdx.x) >> 5;
    int lane = threadIdx.x & 31;
    int lo = lane & 15;
    int h2 = (lane >> 4) << 1;
    int b  = wave / (NK / 16);
    int p0 = (wave % (NK / 16)) * 16;
    float* xb = buf + (size_t)b * 32 * NK;

    v2f alo[8], ahi[8];
#pragma unroll
    for (int s = 0; s < 8; ++s) {
        int kk = 4 * s + h2;
        alo[s].x = W[lo * 32 + kk];        alo[s].y = W[lo * 32 + kk + 1];
        ahi[s].x = W[(lo + 16) * 32 + kk]; ahi[s].y = W[(lo + 16) * 32 + kk + 1];
    }
    v8f clo = {}, chi = {};
#pragma unroll
    for (int s = 0; s < 8; ++s) {
        int kk = 4 * s + h2;
        float x0 = xb[(size_t)kk * NK + p0 + lo];
        float x1 = xb[(size_t)(kk + 1) * NK + p0 + lo];
        float a0 = gnact(x0, st[b * 32 + kk],     gamma[kk],     beta[kk]);
        float a1 = gnact(x1, st[b * 32 + kk + 1], gamma[kk + 1], beta[kk + 1]);
        v2f bf; bf.x = a0; bf.y = a1;
        clo = __builtin_amdgcn_wmma_f32_16x16x4_f32(false, alo[s], false, bf, (short)0, clo, false, false);
        chi = __builtin_amdgcn_wmma_f32_16x16x4_f32(false, ahi[s], false, bf, (short)0, chi, false, false);
    }
    int rb = (lane >= 16) ? 8 : 0;
#pragma unroll
    for (int v = 0; v < 8; ++v) {
        int r0 = v + rb;
        xb[(size_t)r0 * NK + p0 + lo]        = clo[v] + bias[r0];
        xb[(size_t)(r0 + 16) * NK + p0 + lo] = chi[v] + bias[r0 + 16];
    }
}

// -------------------------- SE: s[b][c] = mean over NK of mish(gn3(A1))
__global__ void se_reduce_kernel(const float* __restrict__ A1, const float2* __restrict__ st3,
                                 const float* __restrict__ g3, const float* __restrict__ be3,
                                 float* __restrict__ s_out) {
    int bc = blockIdx.x;
    int c = bc & 31;
    const float4* p = (const float4*)(A1 + (size_t)bc * NK);
    float2 st = st3[bc];
    float g = g3[c], be = be3[c];
    float s = 0.f;
    for (int i = threadIdx.x; i < (NK >> 2); i += 256) {
        float4 v = p[i];
        s += mishf((v.x - st.x) * st.y * g + be);
        s += mishf((v.y - st.x) * st.y * g + be);
        s += mishf((v.z - st.x) * st.y * g + be);
        s += mishf((v.w - st.x) * st.y * g + be);
    }
    __shared__ float sh[256];
    sh[threadIdx.x] = s; __syncthreads();
    for (int w = 128; w > 0; w >>= 1) {
        if (threadIdx.x < w) sh[threadIdx.x] += sh[threadIdx.x + w];
        __syncthreads();
    }
    if (threadIdx.x == 0) s_out[bc] = sh[0] / (float)NK;
}

__global__ void se_gate_kernel(const float* __restrict__ s, const float* __restrict__ se1,
                               const float* __restrict__ se2, float* __restrict__ gate) {
    int t = threadIdx.x;
    if (t >= Bb * 32) return;
    int b = t >> 5; int c = t & 31;
    float h0 = 0.f, h1 = 0.f;
    for (int i = 0; i < 32; ++i) {
        float sv = s[b * 32 + i];
        h0 = fmaf(sv, se1[i * 2 + 0], h0);
        h1 = fmaf(sv, se1[i * 2 + 1], h1);
    }
    h0 = fmaxf(h0, 0.f); h1 = fmaxf(h1, 0.f);
    float g = fmaf(h0, se2[c], h1 * se2[32 + c]);
    gate[t] = 1.f / (1.f + __expf(-g));
}

// -------- conv5: concat[mish(gn3(A1))*gate, mish(gn4(P1))] -> 64x64 GEMM -> F
__global__ void __launch_bounds__(256) conv5_kernel(const float* __restrict__ A1,
        const float* __restrict__ P1, float* __restrict__ F,
        const float2* __restrict__ st3, const float2* __restrict__ st4,
        const float* __restrict__ g3, const float* __restrict__ be3,
        const float* __restrict__ g4, const float* __restrict__ be4,
        const float* __restrict__ gate,
        const float* __restrict__ W5, const float* __restrict__ b5) {
    int wave = (blockIdx.x * 256 + threadIdx.x) >> 5;
    int lane = threadIdx.x & 31;
    int lo = lane & 15;
    int h2 = (lane >> 4) << 1;
    int b  = wave / (NK / 16);
    int p0 = (wave % (NK / 16)) * 16;

    v8f cacc[4] = {};
#pragma unroll
    for (int s = 0; s < 16; ++s) {
        int kk = 4 * s + h2;
        float a0, a1;
        if (kk < 32) {
            float x0 = A1[((size_t)b * 32 + kk) * NK + p0 + lo];
            float x1 = A1[((size_t)b * 32 + kk + 1) * NK + p0 + lo];
            a0 = gnact(x0, st3[b * 32 + kk],     g3[kk],     be3[kk])     * gate[b * 32 + kk];
            a1 = gnact(x1, st3[b * 32 + kk + 1], g3[kk + 1], be3[kk + 1]) * gate[b * 32 + kk + 1];
        } else {
            int q = kk - 32;
            float x0 = P1[((size_t)b * 32 + q) * NK + p0 + lo];
            float x1 = P1[((size_t)b * 32 + q + 1) * NK + p0 + lo];
            a0 = gnact(x0, st4[b * 32 + q],     g4[q],     be4[q]);
            a1 = gnact(x1, st4[b * 32 + q + 1], g4[q + 1], be4[q + 1]);
        }
        v2f bf; bf.x = a0; bf.y = a1;
#pragma unroll
        for (int r = 0; r < 4; ++r) {
            v2f af;
            af.x = W5[(size_t)(r * 16 + lo) * 64 + kk];
            af.y = W5[(size_t)(r * 16 + lo) * 64 + kk + 1];
            cacc[r] = __builtin_amdgcn_wmma_f32_16x16x4_f32(false, af, false, bf, (short)0, cacc[r], false, false);
        }
    }
    int rb = (lane >= 16) ? 8 : 0;
#pragma unroll
    for (int r = 0; r < 4; ++r) {
#pragma unroll
        for (int v = 0; v < 8; ++v) {
            int row = r * 16 + v + rb;
            F[((size_t)b * 64 + row) * NK + p0 + lo] = cacc[r][v] + b5[row];
        }
    }
}

// --------------------- GN5 + mish + max over K -> out (B,64,N)
__global__ void out_kernel(const float* __restrict__ F, const float2* __restrict__ st5,
                           const float* __restrict__ g5, const float* __restrict__ be5,
                           float* __restrict__ out) {
    int t = blockIdx.x * 256 + threadIdx.x;
    if (t >= Bb * 64 * Nn) return;
    int n = t % Nn; int c = (t / Nn) % 64; int b = t / (64 * Nn);
    const float* f = F + ((size_t)b * 64 + c) * NK + (size_t)n * Kk;
    float2 st = st5[b * 32 + (c >> 1)];
    float g = g5[c], be = be5[c];
    float m = -3.0e38f;
#pragma unroll
    for (int k = 0; k < Kk; ++k) m = fmaxf(m, mishf((f[k] - st.x) * st.y * g + be));
    out[t] = m;
}

extern "C" void kernel_launch(void* const* d_in, const int* in_sizes, int n_in,
                              void* d_out, int out_size, void* d_ws, size_t ws_size,
                              hipStream_t stream) {
    const float* x   = (const float*)d_in[0];
    const float* pos = (const float*)d_in[1];
    const float* w1  = (const float*)d_in[2];
    const float* b1  = (const float*)d_in[3];
    const float* g1  = (const float*)d_in[4];
    const float* be1 = (const float*)d_in[5];
    const float* w2  = (const float*)d_in[6];
    const float* b2  = (const float*)d_in[7];
    const float* g2  = (const float*)d_in[8];
    const float* be2 = (const float*)d_in[9];
    const float* w3  = (const float*)d_in[10];
    const float* b3  = (const float*)d_in[11];
    const float* g3  = (const float*)d_in[12];
    const float* be3 = (const float*)d_in[13];
    const float* se1 = (const float*)d_in[14];
    const float* se2 = (const float*)d_in[15];
    const float* w4  = (const float*)d_in[16];
    const float* b4  = (const float*)d_in[17];
    const float* g4  = (const float*)d_in[18];
    const float* be4 = (const float*)d_in[19];
    const float* w5  = (const float*)d_in[20];
    const float* b5  = (const float*)d_in[21];
    const float* g5  = (const float*)d_in[22];
    const float* be5 = (const float*)d_in[23];
    (void)in_sizes; (void)n_in; (void)out_size; (void)ws_size;

    float* out      = (float*)d_out;
    float* dist_sum = out + (size_t)Bb * 64 * Nn;

    char* ws = (char*)d_ws;
    size_t off = 0;
    auto take = [&](size_t bytes) -> char* {
        char* p = ws + off;
        off += (bytes + 255) & ~(size_t)255;
        return p;
    };
    float*  sq   = (float*)take(sizeof(float) * Bb * Nn);
    int*    idx  = (int*)  take(sizeof(int) * (size_t)Bb * Nn * Kk);
    float*  A1   = (float*)take(sizeof(float) * (size_t)Bb * 32 * NK);
    float*  P1   = (float*)take(sizeof(float) * (size_t)Bb * 32 * NK);
    float*  Fb   = (float*)take(sizeof(float) * (size_t)Bb * 64 * NK);
    float2* st1  = (float2*)take(sizeof(float2) * Bb * 32);
    float2* st2  = (float2*)take(sizeof(float2) * Bb * 32);
    float2* st3  = (float2*)take(sizeof(float2) * Bb * 32);
    float2* st4  = (float2*)take(sizeof(float2) * Bb * 32);
    float2* st5  = (float2*)take(sizeof(float2) * Bb * 32);
    float*  sse  = (float*)take(sizeof(float) * Bb * 32);
    float*  gate = (float*)take(sizeof(float) * Bb * 32);

    sq_kernel<<<(Bb * Nn + 255) / 256, 256, 0, stream>>>(pos, sq);
    knn_kernel<<<Bb * (Nn / 16), 32, 0, stream>>>(pos, sq, idx, dist_sum);
    edge_kernel<<<(Bb * Nn * Kk + 255) / 256, 256, 0, stream>>>(x, pos, idx, w1, b1, w2, b2, A1, P1);
    stats_kernel<<<Bb * 32, 256, 0, stream>>>(A1, st1, NK);
    stats_kernel<<<Bb * 32, 256, 0, stream>>>(P1, st2, NK);
    gemm32_kernel<<<(Bb * (NK / 16)) / 8, 256, 0, stream>>>(A1, st1, g1, be1, w3, b3);
    gemm32_kernel<<<(Bb * (NK / 16)) / 8, 256, 0, stream>>>(P1, st2, g2, be2, w4, b4);
    stats_kernel<<<Bb * 32, 256, 0, stream>>>(A1, st3, NK);
    stats_kernel<<<Bb * 32, 256, 0, stream>>>(P1, st4, NK);
    se_reduce_kernel<<<Bb * 32, 256, 0, stream>>>(A1, st3, g3, be3, sse);
    se_gate_kernel<<<1, 128, 0, stream>>>(sse, se1, se2, gate);
    conv5_kernel<<<(Bb * (NK / 16)) / 8, 256, 0, stream>>>(A1, P1, Fb, st3, st4, g3, be3, g4, be4, gate, w5, b5);
    stats_kernel<<<Bb * 32, 256, 0, stream>>>(Fb, st5, 2 * NK);
    out_kernel<<<(Bb * 64 * Nn + 255) / 256, 256, 0, stream>>>(Fb, st5, g5, be5, out);
}